// adder2d_21122649162441
// MI455X (gfx1250) — compile-verified
//
#include <hip/hip_runtime.h>
#include <stdint.h>

// AdderNet conv2d: out[n,f,i,j] = -sum_{c,di,dj} |W[f,c,di,dj] - xpad[n,c,i+di,j+dj]|
// x: [16,64,64,64] f32, W: [64,64,3,3] f32, out: [16,64,64,64] f32. pad=1, stride=1.

#define N_     16
#define CIN    64
#define H_     64
#define W_     64
#define COUT   64
#define TH     16             // tile height
#define TW     32             // tile width (2 pixels per thread along x)
#define HALO_H 18
#define HALO_W 34
#define CIN_T  16             // channel chunk
#define F_T    32             // filters per block
#define NCHUNK (CIN / CIN_T)  // 4

#define SW_ELEMS (9 * CIN_T * F_T)           // 4608 floats, layout [t][cc][f]
#define SX_ELEMS (CIN_T * HALO_H * HALO_W)   // 9792 floats, layout [cc][row][col]
#define XROW    (HALO_H * HALO_W)            // 612

__device__ __forceinline__ void async_b32(unsigned lds, const float* gp) {
    asm volatile("global_load_async_to_lds_b32 %0, %1, off"
                 :: "v"(lds), "v"((uint64_t)(const void*)gp) : "memory");
}

__device__ __forceinline__ void stage_x_row(float* sxb, const float* __restrict__ xg,
                                            int n, int c0, int task,
                                            int h0, int w0) {
    const int rr = task >> 4;   // 0..17
    const int cc = task & 15;   // 0..15
    const int ih = h0 - 1 + rr;
    float* lrow = &sxb[cc * XROW + rr * HALO_W];
    if ((unsigned)ih < (unsigned)H_) {
        const float* g = xg + (((size_t)n * CIN + (c0 + cc)) * H_ + ih) * W_ + (w0 - 1);
#pragma unroll
        for (int col = 0; col < HALO_W; ++col) {
            const int iw = w0 - 1 + col;
            if ((unsigned)iw < (unsigned)W_)
                async_b32((unsigned)(uintptr_t)&lrow[col], g + col);
            else
                lrow[col] = 0.0f;     // ds_store; disjoint from async-written slots
        }
    } else {
#pragma unroll
        for (int col = 0; col < HALO_W; ++col) lrow[col] = 0.0f;
    }
}

__device__ __forceinline__ void stage_chunk(float* swb, float* sxb,
                                            const float* __restrict__ xg,
                                            const float* __restrict__ wg,
                                            int tid, int n, int f0, int c0,
                                            int h0, int w0) {
    // ---- W chunk, LDS layout [t][cc][f] : i = (t*16+cc)*32 + f (shift/mask only)
    for (int i = tid; i < SW_ELEMS; i += 256) {
        const int f  = i & 31;
        const int ct = i >> 5;      // 0..143
        const int t  = ct >> 4;     // 0..8
        const int cc = ct & 15;
        const float* gp = wg + ((size_t)(f0 + f) * CIN + (c0 + cc)) * 9 + t;
        async_b32((unsigned)(uintptr_t)&swb[i], gp);
    }
    // ---- x halo tile: 288 row-tasks (18 rows x 16 channels), 34 cols each
    stage_x_row(sxb, xg, n, c0, tid, h0, w0);               // tasks 0..255
    if (tid < 32)
        stage_x_row(sxb, xg, n, c0, 256 + tid, h0, w0);     // tasks 256..287
}

__global__ __launch_bounds__(256)
void adder2d_kernel(const float* __restrict__ xg,
                    const float* __restrict__ wg,
                    float* __restrict__ outg)
{
    __shared__ __align__(16) float sW[2][SW_ELEMS];   // 2 x 18432 B
    __shared__ __align__(16) float sX[2][SX_ELEMS];   // 2 x 39168 B

    const int tid  = threadIdx.x;
    const int tile = blockIdx.x;            // 0..7 : 4 (h) x 2 (w) tiles
    const int n    = blockIdx.y;            // 0..15
    const int f0   = blockIdx.z * F_T;      // 0 or 32
    const int h0   = (tile >> 1) * TH;
    const int w0   = (tile & 1)  * TW;
    const int ty   = tid >> 4;              // 0..15
    const int tx   = tid & 15;              // 0..15 ; pixels (ty,tx) and (ty,tx+16)

    float acc0[F_T], acc1[F_T];
#pragma unroll
    for (int f = 0; f < F_T; ++f) { acc0[f] = 0.0f; acc1[f] = 0.0f; }

    // ---- prologue: stage chunk 0 into buffer 0
    stage_chunk(sW[0], sX[0], xg, wg, tid, n, f0, 0, h0, w0);
    asm volatile("s_wait_asynccnt 0x0" ::: "memory");
    __syncthreads();

    int cur = 0;
    for (int i = 0; i < NCHUNK; ++i) {
        // ---- prefetch next chunk into the other buffer (runs under compute)
        if (i + 1 < NCHUNK)
            stage_chunk(sW[cur ^ 1], sX[cur ^ 1], xg, wg, tid, n, f0,
                        (i + 1) * CIN_T, h0, w0);

        // ---- compute on current buffer: 2 pixels per thread share each W load
        const float* swb = sW[cur];
        const float* sxb = sX[cur];
        for (int cc = 0; cc < CIN_T; ++cc) {
            const float* px = &sxb[cc * XROW + ty * HALO_W + tx];
            float p0[9], p1[9];
#pragma unroll
            for (int r = 0; r < 3; ++r) {
#pragma unroll
                for (int c = 0; c < 3; ++c) {
                    p0[r * 3 + c] = px[r * HALO_W + c];
                    p1[r * 3 + c] = px[r * HALO_W + c + 16];
                }
            }
#pragma unroll
            for (int t = 0; t < 9; ++t) {
                const float4* wq = (const float4*)&swb[(t * CIN_T + cc) * F_T];
#pragma unroll
                for (int q = 0; q < 8; ++q) {         // broadcast ds_load_b128
                    const float4 w4 = wq[q];
                    acc0[4 * q + 0] += fabsf(w4.x - p0[t]);
                    acc0[4 * q + 1] += fabsf(w4.y - p0[t]);
                    acc0[4 * q + 2] += fabsf(w4.z - p0[t]);
                    acc0[4 * q + 3] += fabsf(w4.w - p0[t]);
                    acc1[4 * q + 0] += fabsf(w4.x - p1[t]);
                    acc1[4 * q + 1] += fabsf(w4.y - p1[t]);
                    acc1[4 * q + 2] += fabsf(w4.z - p1[t]);
                    acc1[4 * q + 3] += fabsf(w4.w - p1[t]);
                }
            }
        }

        asm volatile("s_wait_asynccnt 0x0" ::: "memory");
        __syncthreads();
        cur ^= 1;
    }

    const int oh  = h0 + ty;
    const int ow0 = w0 + tx;
#pragma unroll
    for (int f = 0; f < F_T; ++f) {
        float* orow = outg + (((size_t)n * COUT + (f0 + f)) * H_ + oh) * W_;
        orow[ow0]      = -acc0[f];
        orow[ow0 + 16] = -acc1[f];
    }
}

extern "C" void kernel_launch(void* const* d_in, const int* in_sizes, int n_in,
                              void* d_out, int out_size, void* d_ws, size_t ws_size,
                              hipStream_t stream) {
    const float* x = (const float*)d_in[0];   // [16,64,64,64]
    const float* w = (const float*)d_in[1];   // [64,64,3,3]
    float* out = (float*)d_out;               // [16,64,64,64]
    (void)in_sizes; (void)n_in; (void)out_size; (void)d_ws; (void)ws_size;

    dim3 grid(8, N_, COUT / F_T);   // 4x2 tiles x 16 images x 2 filter groups
    adder2d_kernel<<<grid, 256, 0, stream>>>(x, w, out);
}